// FreeLoss_70265664963107
// MI455X (gfx1250) — compile-verified
//
#include <hip/hip_runtime.h>
#include <stdint.h>

#define BB 8
#define AA 65536
#define CC 80
#define GG 16
#define KK 50
#define VAR0 0.1f
#define VAR1 0.2f
#define BBOX_T 0.5f
#define SL1_W 0.75f
#define SL1_BETA 0.11f
#define ALPHA 0.5f

typedef __attribute__((ext_vector_type(2))) float v2f;
typedef __attribute__((ext_vector_type(8))) float v8f;

__device__ __forceinline__ float sigmoidf_(float x){ return 1.0f/(1.0f + expf(-x)); }

__device__ __forceinline__ float iou_corner(float ax1,float ay1,float ax2,float ay2,
                                            float bx1,float by1,float bx2,float by2){
  float iw = fmaxf(fminf(ax2,bx2) - fmaxf(ax1,bx1), 0.0f);
  float ih = fmaxf(fminf(ay2,by2) - fmaxf(ay1,by1), 0.0f);
  float inter = iw*ih;
  float aa = (ax2-ax1)*(ay2-ay1);
  float ab = (bx2-bx1)*(by2-by1);
  return inter / (aa + ab - inter);
}

__device__ __forceinline__ float sl1_(float v){
  float av = fabsf(v);
  return (av < SL1_BETA) ? (0.5f/SL1_BETA)*v*v : (av - 0.5f*SL1_BETA);
}

// ---------------- kernel 1: t2[b,g] = max_a iou(tbox, decoded box), clipped at 0.5
__global__ void k_t2(const float* __restrict__ breg, const float* __restrict__ anc,
                     const float* __restrict__ tbox, float* __restrict__ t2){
  int g = blockIdx.x, b = blockIdx.y;
  const float4 tb = ((const float4*)tbox)[b*GG + g];
  float m = 0.0f;
  for (int a = threadIdx.x; a < AA; a += 256){
    float4 br = ((const float4*)breg)[(size_t)b*AA + a];
    float4 pr = ((const float4*)anc)[a];
    float cx = pr.x + br.x*VAR0*pr.z;
    float cy = pr.y + br.y*VAR0*pr.w;
    float w  = pr.z * expf(br.z*VAR1);
    float h  = pr.w * expf(br.w*VAR1);
    float iou = iou_corner(cx-0.5f*w, cy-0.5f*h, cx+0.5f*w, cy+0.5f*h,
                           tb.x, tb.y, tb.z, tb.w);
    m = fmaxf(m, iou);
  }
  __shared__ float sm[256];
  sm[threadIdx.x] = m; __syncthreads();
  for (int s = 128; s > 0; s >>= 1){
    if (threadIdx.x < s) sm[threadIdx.x] = fmaxf(sm[threadIdx.x], sm[threadIdx.x+s]);
    __syncthreads();
  }
  if (threadIdx.x == 0) t2[b*GG+g] = fmaxf(sm[0], BBOX_T);  // +1e-12 vanishes in f32
}

// ---------------- kernel 2: top-50 anchors per (b,g) by mqm, full row in 256KB LDS
__global__ void k_topk(const float* __restrict__ anc, const float* __restrict__ tbox,
                       int* __restrict__ matched){
  extern __shared__ float mq[];                    // AA floats (256 KB)
  __shared__ float sv[2048]; __shared__ int si[2048];
  __shared__ float rv[16];   __shared__ int ri[16];
  int tid = threadIdx.x, g = blockIdx.x, b = blockIdx.y;
  const float4 tb = ((const float4*)tbox)[b*GG + g];
  for (int a = tid; a < AA; a += 512){
    float4 pr = ((const float4*)anc)[a];
    mq[a] = iou_corner(pr.x-0.5f*pr.z, pr.y-0.5f*pr.w, pr.x+0.5f*pr.z, pr.y+0.5f*pr.w,
                       tb.x, tb.y, tb.z, tb.w);
  }
  __syncthreads();
  for (int j = 0; j < 4; ++j){
    int s = tid + j*512;
    float bv = -2.0f; int bi = -1;
    for (int i = 0; i < 32; ++i){ float v = mq[s*32+i]; if (v > bv){ bv = v; bi = s*32+i; } }
    sv[s] = bv; si[s] = bi;
  }
  __syncthreads();
  int lane = tid & 31, wave = tid >> 5;
  for (int k = 0; k < KK; ++k){
    float bv = -2.0f; int bi = 1 << 30;
    for (int j = 0; j < 4; ++j){
      int s = tid + j*512;
      float v = sv[s]; int ii = si[s];
      if (v > bv || (v == bv && ii < bi)){ bv = v; bi = ii; }
    }
    for (int off = 16; off > 0; off >>= 1){
      float ov = __shfl_down(bv, off, 32);
      int   oi = __shfl_down(bi, off, 32);
      if (ov > bv || (ov == bv && oi < bi)){ bv = ov; bi = oi; }
    }
    if (lane == 0){ rv[wave] = bv; ri[wave] = bi; }
    __syncthreads();
    if (tid == 0){
      float fv = rv[0]; int fi = ri[0];
      for (int w2 = 1; w2 < 16; ++w2)
        if (rv[w2] > fv || (rv[w2] == fv && ri[w2] < fi)){ fv = rv[w2]; fi = ri[w2]; }
      matched[(b*GG+g)*KK + k] = fi;
      mq[fi] = -1.0f;
      int s = fi >> 5;
      float nbv = -2.0f; int nbi = -1;
      for (int i = 0; i < 32; ++i){ float v = mq[s*32+i]; if (v > nbv){ nbv = v; nbi = s*32+i; } }
      sv[s] = nbv; si[s] = nbi;
    }
    __syncthreads();
  }
}

// ---------------- kernel 3: positive bag loss, one thread per (b,g)
__global__ void k_pos(const float* __restrict__ breg, const float* __restrict__ cls,
                      const float* __restrict__ anc, const float* __restrict__ tbox,
                      const int* __restrict__ lab, const int* __restrict__ matched,
                      float* __restrict__ pos){
  int t = threadIdx.x;                // 0..127
  int b = t / GG, g = t % GG;
  float4 tb = ((const float4*)tbox)[b*GG + g];
  int lb = lab[b*GG + g];
  float mcx = 0.5f*(tb.x + tb.z), mcy = 0.5f*(tb.y + tb.w);
  float mw = tb.z - tb.x, mh = tb.w - tb.y;
  float sw = 0.0f, swz = 0.0f;
  for (int k = 0; k < KK; ++k){
    int j = matched[(b*GG+g)*KK + k];
    float p  = sigmoidf_(cls[((size_t)b*AA + j)*CC + lb]);
    float4 pr = ((const float4*)anc)[j];
    float4 br = ((const float4*)breg)[(size_t)b*AA + j];
    float gx = (mcx - pr.x) / (VAR0*pr.z);
    float gy = (mcy - pr.y) / (VAR0*pr.w);
    float gw = logf(mw/pr.z) / VAR1;
    float gh = logf(mh/pr.w) / VAR1;
    float rl = SL1_W*(sl1_(gx-br.x)+sl1_(gy-br.y)+sl1_(gw-br.z)+sl1_(gh-br.w));
    float z = p * expf(-rl);
    float w = 1.0f / fmaxf(1.0f - z, 1e-12f);
    sw += w; swz += w*z;
  }
  pos[t] = -logf(swz / sw);
}

// ---------------- kernel 4: negative loss (streams cls_logits, 168 MB)
__global__ void k_neg(const float* __restrict__ breg, const float* __restrict__ cls,
                      const float* __restrict__ anc, const float* __restrict__ tbox,
                      const int* __restrict__ lab, const float* __restrict__ t2,
                      float* __restrict__ negp){
  __shared__ float4 stb[GG];
  __shared__ int    sll[GG];
  __shared__ float  sdn[GG];
  __shared__ float4 sbr[256];
  __shared__ float4 san[256];
  __shared__ float  wred[16];
  int tid = threadIdx.x;
  int b = blockIdx.y;
  int a0 = blockIdx.x * 256;
  if (tid < GG){
    stb[tid] = ((const float4*)tbox)[b*GG + tid];
    sll[tid] = lab[b*GG + tid];
    sdn[tid] = t2[b*GG + tid] - BBOX_T;
  }
  // CDNA5 async copy: stage this block's box_reg & anchor tiles into LDS (ASYNCcnt path)
  {
    uint32_t ldsb = (uint32_t)(uintptr_t)(&sbr[tid]);
    uint32_t ldsa = (uint32_t)(uintptr_t)(&san[tid]);
    uint32_t voff = (uint32_t)((a0 + tid) * 16u);
    uint64_t gb = (uint64_t)(uintptr_t)breg + (uint64_t)b * AA * 16u;
    uint64_t ga = (uint64_t)(uintptr_t)anc;
    asm volatile("global_load_async_to_lds_b128 %0, %1, %2"
                 :: "v"(ldsb), "v"(voff), "s"(gb) : "memory");
    asm volatile("global_load_async_to_lds_b128 %0, %1, %2"
                 :: "v"(ldsa), "v"(voff), "s"(ga) : "memory");
    asm volatile("s_wait_asynccnt 0" ::: "memory");
  }
  __syncthreads();
  int a = a0 + tid;
  float4 br = sbr[tid];
  float4 pr = san[tid];
  float cx = pr.x + br.x*VAR0*pr.z;
  float cy = pr.y + br.y*VAR0*pr.w;
  float w  = pr.z * expf(br.z*VAR1);
  float h  = pr.w * expf(br.w*VAR1);
  float dx1 = cx-0.5f*w, dy1 = cy-0.5f*h, dx2 = cx+0.5f*w, dy2 = cy+0.5f*h;
  float areaD = (dx2-dx1)*(dy2-dy1);
  float ob[GG];
  #pragma unroll
  for (int g = 0; g < GG; ++g){
    float4 t = stb[g];
    float iw = fmaxf(fminf(dx2,t.z) - fmaxf(dx1,t.x), 0.0f);
    float ih = fmaxf(fminf(dy2,t.w) - fmaxf(dy1,t.y), 0.0f);
    float inter = iw*ih;
    float at = (t.z-t.x)*(t.w-t.y);
    float iou = inter / (at + areaD - inter);
    ob[g] = fminf(fmaxf((iou - BBOX_T)/sdn[g], 0.0f), 1.0f);
  }
  const float4* lrow = (const float4*)(cls + ((size_t)b*AA + a)*CC);
  float acc = 0.0f;
  #pragma unroll 4
  for (int q = 0; q < CC/4; ++q){
    float4 l = lrow[q];
    float p0 = sigmoidf_(l.x); acc += p0*p0*(-log1pf(-p0));
    float p1 = sigmoidf_(l.y); acc += p1*p1*(-log1pf(-p1));
    float p2 = sigmoidf_(l.z); acc += p2*p2*(-log1pf(-p2));
    float p3 = sigmoidf_(l.w); acc += p3*p3*(-log1pf(-p3));
  }
  // corrections for the <=16 classes with nonzero box_prob (dedup by first occurrence)
  #pragma unroll
  for (int g = 0; g < GG; ++g){
    int c = sll[g];
    bool first = true;
    #pragma unroll
    for (int g2 = 0; g2 < g; ++g2) if (sll[g2] == c) first = false;
    float bp = ob[g];
    #pragma unroll
    for (int g2 = g+1; g2 < GG; ++g2) if (sll[g2] == c) bp = fmaxf(bp, ob[g2]);
    if (first && bp > 0.0f){
      float p = sigmoidf_(cls[((size_t)b*AA + a)*CC + c]);
      float x = p*(1.0f - bp);
      acc += x*x*(-log1pf(-x)) - p*p*(-log1pf(-p));
    }
  }
  // wave-level sum via WMMA f32 16x16x4 with all-ones B: D[m][n] = rowsum_m
  int lane = tid & 31, wave = tid >> 5;
#if __has_builtin(__builtin_amdgcn_wmma_f32_16x16x4_f32)
  v2f aM; aM[0] = acc;  aM[1] = 0.0f;
  v2f bM; bM[0] = 1.0f; bM[1] = 1.0f;
  v8f cM = {0.f,0.f,0.f,0.f,0.f,0.f,0.f,0.f};
  v8f dM = __builtin_amdgcn_wmma_f32_16x16x4_f32(false, aM, false, bM,
                                                 (short)0, cM, false, false);
  float s8 = dM[0]+dM[1]+dM[2]+dM[3]+dM[4]+dM[5]+dM[6]+dM[7];
  if (lane == 0)  wred[wave*2+0] = s8;   // sum of lanes 0..7, 16..23
  if (lane == 16) wred[wave*2+1] = s8;   // sum of lanes 8..15, 24..31
#else
  float s = acc;
  for (int off = 16; off > 0; off >>= 1) s += __shfl_down(s, off, 32);
  if (lane == 0){ wred[wave*2+0] = s; wred[wave*2+1] = 0.0f; }
#endif
  __syncthreads();
  if (tid == 0){
    float s = 0.0f;
    #pragma unroll
    for (int i = 0; i < 16; ++i) s += wred[i];
    negp[blockIdx.y*gridDim.x + blockIdx.x] = s;   // deterministic per-block partial
  }
}

// ---------------- kernel 5: deterministic final reduction + scaling
__global__ void k_final(const float* __restrict__ pos, const float* __restrict__ negp,
                        float* __restrict__ out){
  __shared__ float sm[256];
  int t = threadIdx.x;
  float s = 0.0f;
  for (int i = t; i < 2048; i += 256) s += negp[i];
  sm[t] = s; __syncthreads();
  for (int st = 128; st > 0; st >>= 1){ if (t < st) sm[t] += sm[t+st]; __syncthreads(); }
  float negsum = sm[0];
  __syncthreads();
  sm[t] = (t < 128) ? pos[t] : 0.0f; __syncthreads();
  for (int st = 128; st > 0; st >>= 1){ if (t < st) sm[t] += sm[t+st]; __syncthreads(); }
  if (t == 0){
    out[0] = ALPHA * sm[0] / (float)(BB*GG);
    out[1] = (1.0f - ALPHA) * negsum / (float)(BB*GG*KK);
  }
}

extern "C" void kernel_launch(void* const* d_in, const int* in_sizes, int n_in,
                              void* d_out, int out_size, void* d_ws, size_t ws_size,
                              hipStream_t stream){
  (void)in_sizes; (void)n_in; (void)out_size; (void)ws_size;
  const float* breg = (const float*)d_in[0];
  const float* cls  = (const float*)d_in[1];
  const float* anc  = (const float*)d_in[2];
  const float* tbox = (const float*)d_in[3];
  const int*   lab  = (const int*)d_in[4];
  float* out = (float*)d_out;

  float* t2      = (float*)d_ws;                 // 128 f32
  int*   matched = (int*)(t2 + BB*GG);           // 6400 i32
  float* pos     = (float*)(matched + BB*GG*KK); // 128 f32
  float* negp    = pos + BB*GG;                  // 2048 f32

  k_t2  <<<dim3(GG, BB), 256, 0, stream>>>(breg, anc, tbox, t2);
  k_topk<<<dim3(GG, BB), 512, AA*sizeof(float), stream>>>(anc, tbox, matched);
  k_pos <<<1, 128, 0, stream>>>(breg, cls, anc, tbox, lab, matched, pos);
  k_neg <<<dim3(AA/256, BB), 256, 0, stream>>>(breg, cls, anc, tbox, lab, t2, negp);
  k_final<<<1, 256, 0, stream>>>(pos, negp, out);
}